// RankModelF_39273180954756
// MI455X (gfx1250) — compile-verified
//
#include <hip/hip_runtime.h>
#include <stdint.h>

typedef __attribute__((ext_vector_type(2))) float v2f;
typedef __attribute__((ext_vector_type(4))) float v4f;
typedef __attribute__((ext_vector_type(8))) float v8f;
typedef __attribute__((ext_vector_type(4))) int   v4i;

#define RPB   256   // rows (batch elements) per block
#define NCOL  9     // 1 query + 8 refs
#define NOUT  56    // permutations(8,2)
#define SROW  9     // padded LDS row stride for s/c arrays (bank-conflict free)

// ---- gfx1250 async global->LDS staging (unguarded: decisive compile probe) ----
// Block tile = validRows*36 bytes, contiguous; base is 16B-aligned for full blocks.
__device__ __forceinline__ void stage_stim(const int* gsrc, int* ldst, int nInts, int t)
{
    typedef __attribute__((address_space(1))) v4i g4_t;
    typedef __attribute__((address_space(3))) v4i l4_t;
    typedef __attribute__((address_space(1))) int g1_t;
    typedef __attribute__((address_space(3))) int l1_t;
    g4_t* g4 = (g4_t*)(uintptr_t)gsrc;
    l4_t* l4 = (l4_t*)(uintptr_t)ldst;
    const int nInt4 = nInts >> 2;
    for (int f = t; f < nInt4; f += RPB)
        __builtin_amdgcn_global_load_async_to_lds_b128(g4 + f, l4 + f, 0, 0);
    const int rem = nInts & 3;
    if (t < rem) {
        g1_t* g1 = (g1_t*)(uintptr_t)gsrc;
        l1_t* l1 = (l1_t*)(uintptr_t)ldst;
        const int f = (nInt4 << 2) + t;
        __builtin_amdgcn_global_load_async_to_lds_b32(g1 + f, l1 + f, 0, 0);
    }
}

__device__ __forceinline__ void wait_stage()
{
#if __has_builtin(__builtin_amdgcn_s_wait_asynccnt)
    __builtin_amdgcn_s_wait_asynccnt(0);
#else
    asm volatile("s_wait_asynccnt 0" ::: "memory");
#endif
}

__global__ __launch_bounds__(RPB) void rankprob_kernel(
    const int*  __restrict__ stim,   // (BATCH, 9) int32, values in [1,20]
    const float* __restrict__ emb,   // (21, 3) f32
    float*      __restrict__ out,    // (BATCH, 56) f32
    int batch)
{
    __shared__ float Epad[32 * 4];                   // embedding padded to 32x4
    __shared__ float nrm[32];                        // squared norms
    __shared__ float sTab[32 * 32];                  // similarity table s(q,r)
    __shared__ __align__(16) int stimLds[RPB * NCOL];// staged stimulus rows
    __shared__ float sArr[RPB * SROW];               // per-row s_j
    __shared__ float cArr[RPB * SROW];               // per-row c_i

    const int t         = threadIdx.x;
    const int blockRow0 = blockIdx.x * RPB;
    const int validRows = (batch - blockRow0 < RPB) ? (batch - blockRow0) : RPB;
    const int nInts     = validRows * NCOL;

    // Kick off async staging of this block's stimulus rows (overlapped w/ table build)
    stage_stim(stim + (size_t)blockRow0 * NCOL, stimLds, nInts, t);

    // --- build padded embedding + norms ---
    if (t < 32) {
        float e0 = 0.f, e1 = 0.f, e2 = 0.f;
        if (t < 21) { e0 = emb[t*3+0]; e1 = emb[t*3+1]; e2 = emb[t*3+2]; }
        Epad[t*4+0] = e0; Epad[t*4+1] = e1; Epad[t*4+2] = e2; Epad[t*4+3] = 0.f;
        nrm[t] = e0*e0 + e1*e1 + e2*e2;
    }
    __syncthreads();

    // --- Gram matrix G = E*E^T via V_WMMA_F32_16X16X4_F32, 2x2 tiles of 16 ---
    const int wave = t >> 5;
    const int lane = t & 31;
    if (wave < 4) {
        const int tM   = wave >> 1, tN = wave & 1;
        const int half = lane >> 4;          // 0: lanes 0-15, 1: lanes 16-31
        const int ln   = lane & 15;
        const int kb   = half * 2;           // K pair held by this half-wave
        const int mA   = tM * 16 + ln;
        const int nB   = tN * 16 + ln;

        v2f a, b;
        a.x = Epad[mA*4 + kb]; a.y = Epad[mA*4 + kb + 1];
        b.x = Epad[nB*4 + kb]; b.y = Epad[nB*4 + kb + 1];
        v8f c = {};
        c = __builtin_amdgcn_wmma_f32_16x16x4_f32(
                /*neg_a=*/false, a, /*neg_b=*/false, b,
                /*c_mod=*/(short)0, c, /*reuse_a=*/false, /*reuse_b=*/false);

        // C/D layout: VGPR r, lanes 0-15 -> M=r, lanes 16-31 -> M=r+8; N=lane&15
        #pragma unroll
        for (int r = 0; r < 8; ++r) {
            int mm = tM*16 + r + 8*half;
            int nn = tN*16 + ln;
            float d2 = nrm[mm] + nrm[nn] - 2.0f * c[r];
            float d  = sqrtf(fmaxf(d2, 0.0f));
            sTab[mm*32 + nn] = __expf(-10.0f * d) + 0.001f;
        }
    }

    wait_stage();        // stimLds ready (per-wave async count)
    __syncthreads();     // sTab + stimLds visible to all

    // --- per-row: gather 8 similarities, total, and c_i terms ---
    if (t < validRows) {
        const int* row = &stimLds[t * NCOL];
        const int  q   = row[0];
        float s[8], total = 0.f;
        #pragma unroll
        for (int j = 0; j < 8; ++j) {
            float v = sTab[q*32 + row[1 + j]];
            s[j] = v; total += v;
        }
        #pragma unroll
        for (int j = 0; j < 8; ++j) {
            sArr[t*SROW + j] = s[j];
            cArr[t*SROW + j] = s[j] / (total * (total - s[j]));
        }
    }
    __syncthreads();

    // --- fully coalesced, non-temporal float4 output (write-once stream) ---
    float* outBase = out + (size_t)blockRow0 * NOUT;
    const int limit = validRows * NOUT;
    #pragma unroll
    for (int it = 0; it < 14; ++it) {
        int k = (t + RPB * it) * 4;               // flat float index in block tile
        if (k < limit) {
            int row = k / NOUT;
            int o   = k - row * NOUT;
            const float* sR = &sArr[row * SROW];
            const float* cR = &cArr[row * SROW];
            v4f v;
            {
                int i = (o+0)/7, jj = (o+0)-i*7, j = jj + (jj >= i);
                v.x = cR[i] * sR[j];
            }
            {
                int i = (o+1)/7, jj = (o+1)-i*7, j = jj + (jj >= i);
                v.y = cR[i] * sR[j];
            }
            {
                int i = (o+2)/7, jj = (o+2)-i*7, j = jj + (jj >= i);
                v.z = cR[i] * sR[j];
            }
            {
                int i = (o+3)/7, jj = (o+3)-i*7, j = jj + (jj >= i);
                v.w = cR[i] * sR[j];
            }
            __builtin_nontemporal_store(v, (v4f*)(outBase + k)); // b128 NT, coalesced
        }
    }
}

extern "C" void kernel_launch(void* const* d_in, const int* in_sizes, int n_in,
                              void* d_out, int out_size, void* d_ws, size_t ws_size,
                              hipStream_t stream) {
    (void)n_in; (void)out_size; (void)d_ws; (void)ws_size;
    const int*   stim = (const int*)d_in[0];
    const float* emb  = (const float*)d_in[1];
    float*       out  = (float*)d_out;
    const int batch   = in_sizes[0] / NCOL;
    if (batch <= 0) return;
    const int blocks  = (batch + RPB - 1) / RPB;
    rankprob_kernel<<<blocks, RPB, 0, stream>>>(stim, emb, out, batch);
}